// SetPool_71253507441381
// MI455X (gfx1250) — compile-verified
//
#include <hip/hip_runtime.h>
#include <math.h>

typedef __attribute__((ext_vector_type(2))) float v2f;
typedef __attribute__((ext_vector_type(8))) float v8f;

#define D_DIM 256
#define K1_WAVES 2           // waves per block in logits kernel
#define ROW_PAD 260          // 260*4 B row stride: 16B aligned, bank-conflict free

// ---------------------------------------------------------------------------
// Kernel 1: logits[e] = dot(z[idx[e]], attn_w) + attn_b  via V_WMMA_F32_16X16X4_F32
// One wave per 16-element tile. Rows gathered coalesced into LDS, then 64 WMMA
// steps accumulate the 16 dots (B = attn_w chunk broadcast across all N cols).
// ---------------------------------------------------------------------------
__global__ void __launch_bounds__(K1_WAVES * 32)
logits_wmma_kernel(const float* __restrict__ z,
                   const float* __restrict__ attn_w,
                   const float* __restrict__ attn_b,
                   const int*   __restrict__ idx,
                   float*       __restrict__ logits,
                   int E, int n_tiles, int tile_stride)
{
    __shared__ __align__(16) float rows[K1_WAVES][16][ROW_PAD];
    __shared__ __align__(16) float wvec[D_DIM];

    const int lane = threadIdx.x & 31;
    const int wave = threadIdx.x >> 5;

    // Stage attn_w into LDS (every wave writes identical data; a wave's own
    // LDS writes are in-order with its own later LDS reads).
    for (int j = lane; j < D_DIM; j += 32) wvec[j] = attn_w[j];

    const float bias = *attn_b;

    const int mrow = lane & 15;      // M index within tile for A fragment
    const int hi   = lane >> 4;      // 0: K pair {0,1}, 1: K pair {2,3}

    for (int tile = blockIdx.x * K1_WAVES + wave; tile < n_tiles; tile += tile_stride) {
        const int ebase = tile * 16;

        // ---- coalesced gather of 16 rows into this wave's LDS tile ----
        for (int r = 0; r < 16; ++r) {
            int e = ebase + r; if (e >= E) e = E - 1;           // clamp tail (stores masked later)
            const float4* src = (const float4*)(z + (size_t)idx[e] * D_DIM);
            float4 a0 = src[lane];
            float4 a1 = src[lane + 32];
            float4* dst = (float4*)(&rows[wave][r][0]);
            dst[lane]      = a0;
            dst[lane + 32] = a1;
        }

        // ---- 64 x v_wmma_f32_16x16x4_f32, accumulating C ----
        const float* arow = &rows[wave][mrow][0];
        v8f c = {};
        #pragma unroll 8
        for (int kk = 0; kk < D_DIM; kk += 4) {
            const int k2 = kk + 2 * hi;
            v2f a; a.x = arow[k2]; a.y = arow[k2 + 1];   // A: 16x4 f32 fragment
            v2f b; b.x = wvec[k2]; b.y = wvec[k2 + 1];   // B: 4x16, broadcast over N
            c = __builtin_amdgcn_wmma_f32_16x16x4_f32(
                    /*neg_a=*/false, a, /*neg_b=*/false, b,
                    /*c_mod=*/(short)0, c, /*reuse_a=*/false, /*reuse_b=*/false);
        }

        // C layout: VGPR r, lanes 0-15 -> (M=r, N=lane); lanes 16-31 -> (M=8+r).
        // N column is irrelevant (B broadcast) -> lanes 0 and 16 hold all 16 logits.
        if (mrow == 0) {
            const int lb = ebase + 8 * hi;                 // 32B-aligned
            if (lb + 8 <= E) {                             // fast path: full tile
                float4 lo4, hi4;
                lo4.x = c[0] + bias; lo4.y = c[1] + bias;
                lo4.z = c[2] + bias; lo4.w = c[3] + bias;
                hi4.x = c[4] + bias; hi4.y = c[5] + bias;
                hi4.z = c[6] + bias; hi4.w = c[7] + bias;
                float4* dst = (float4*)(logits + lb);
                dst[0] = lo4;
                dst[1] = hi4;
            } else {                                       // tail tile (at most one)
                for (int r = 0; r < 8; ++r) {
                    const int e = lb + r;
                    if (e < E) logits[e] = c[r] + bias;
                }
            }
        }
    }
}

// ---------------------------------------------------------------------------
// Kernel 2: one block (256 thr = 8 waves) per segment.
//   pass A: block max of logits over [start,end)
//   pass B: per-wave private accumulation of sum(exp(l-m) * row), fixed-order
//           cross-wave merge in LDS (deterministic, no float atomics)
// ---------------------------------------------------------------------------
__global__ void __launch_bounds__(256)
segpool_kernel(const float* __restrict__ z,
               const int*   __restrict__ idx,
               const int*   __restrict__ seg,
               const float* __restrict__ logits,
               float*       __restrict__ out,
               int E)
{
    const int s    = blockIdx.x;
    const int tid  = threadIdx.x;
    const int lane = tid & 31;
    const int wave = tid >> 5;

    // lower_bound(seg, s) and lower_bound(seg, s+1): uniform per block.
    int lo = 0, hb = E;
    while (lo < hb) { int mid = (lo + hb) >> 1; if (seg[mid] < s)     lo = mid + 1; else hb = mid; }
    const int start = lo;
    hb = E;
    while (lo < hb) { int mid = (lo + hb) >> 1; if (seg[mid] < s + 1) lo = mid + 1; else hb = mid; }
    const int end = lo;
    const int len = end - start;

    __shared__ float red[256];
    __shared__ float accs[8][D_DIM];
    __shared__ float denw[8];

    if (len == 0) {                       // empty segment -> zeros (matches ref)
        out[(size_t)s * D_DIM + tid] = 0.0f;
        return;
    }

    // ---- pass A: segment max of logits ----
    float m = -INFINITY;
    for (int e = start + tid; e < end; e += 256) m = fmaxf(m, logits[e]);
    red[tid] = m;
    __syncthreads();
    for (int off = 128; off > 0; off >>= 1) {
        if (tid < off) red[tid] = fmaxf(red[tid], red[tid + off]);
        __syncthreads();
    }
    m = red[0];

    // ---- pass B: wave-per-element weighted accumulation ----
    float a00 = 0.f, a01 = 0.f, a02 = 0.f, a03 = 0.f;
    float a10 = 0.f, a11 = 0.f, a12 = 0.f, a13 = 0.f;
    float den = 0.f;
    for (int e = start + wave; e < end; e += 8) {
        const float t = expf(logits[e] - m);         // uniform across wave
        const float4* src = (const float4*)(z + (size_t)idx[e] * D_DIM);
        const float4 v0 = src[lane];                 // cols 4*lane .. 4*lane+3
        const float4 v1 = src[lane + 32];            // cols 128+4*lane ..
        a00 += t * v0.x; a01 += t * v0.y; a02 += t * v0.z; a03 += t * v0.w;
        a10 += t * v1.x; a11 += t * v1.y; a12 += t * v1.z; a13 += t * v1.w;
        den += t;
    }
    accs[wave][4 * lane + 0]       = a00;
    accs[wave][4 * lane + 1]       = a01;
    accs[wave][4 * lane + 2]       = a02;
    accs[wave][4 * lane + 3]       = a03;
    accs[wave][128 + 4 * lane + 0] = a10;
    accs[wave][128 + 4 * lane + 1] = a11;
    accs[wave][128 + 4 * lane + 2] = a12;
    accs[wave][128 + 4 * lane + 3] = a13;
    if (lane == 0) denw[wave] = den;
    __syncthreads();

    // ---- fixed-order merge (deterministic) ----
    float acc = 0.f, d = 0.f;
    #pragma unroll
    for (int w = 0; w < 8; ++w) { acc += accs[w][tid]; d += denw[w]; }
    out[(size_t)s * D_DIM + tid] = acc / d;          // d >= 1 for non-empty segments
}

// ---------------------------------------------------------------------------
extern "C" void kernel_launch(void* const* d_in, const int* in_sizes, int n_in,
                              void* d_out, int out_size, void* d_ws, size_t ws_size,
                              hipStream_t stream)
{
    const float* z      = (const float*)d_in[0];   // [N, 256]
    const float* attn_w = (const float*)d_in[1];   // [256]
    const float* attn_b = (const float*)d_in[2];   // scalar
    const int*   idx    = (const int*)d_in[3];     // [E]
    const int*   seg    = (const int*)d_in[4];     // [E], sorted
    (void)n_in; (void)ws_size;

    const int E = in_sizes[3];
    const int S = out_size / D_DIM;

    float* logits = (float*)d_ws;                  // E floats of scratch

    const int n_tiles = (E + 15) / 16;
    int blocks1 = (n_tiles + K1_WAVES - 1) / K1_WAVES;
    if (blocks1 > 8192) blocks1 = 8192;
    const int tile_stride = blocks1 * K1_WAVES;

    logits_wmma_kernel<<<blocks1, K1_WAVES * 32, 0, stream>>>(
        z, attn_w, attn_b, idx, logits, E, n_tiles, tile_stride);

    segpool_kernel<<<S, 256, 0, stream>>>(z, idx, seg, logits, (float*)d_out, E);
}